// GATEncoder_28226525069350
// MI455X (gfx1250) — compile-verified
//
#include <hip/hip_runtime.h>
#include <math.h>

#define NEG_SLOPE 0.2f

typedef __attribute__((ext_vector_type(2))) float v2f;
typedef __attribute__((ext_vector_type(8))) float v8f;

// ---------- helpers ----------
__device__ __forceinline__ float lrelu(float v) { return v > 0.f ? v : NEG_SLOPE * v; }

// monotone float <-> uint encoding for atomic segment-max
__device__ __forceinline__ unsigned ford(float f) {
    unsigned u = __float_as_uint(f);
    return (u & 0x80000000u) ? ~u : (u | 0x80000000u);
}
__device__ __forceinline__ float forddec(unsigned u) {
    return (u & 0x80000000u) ? __uint_as_float(u & 0x7fffffffu) : __uint_as_float(~u);
}
#define FORD_NEG_INF 0x007FFFFFu  // ford(-inf)

// ---------- init (ws is poisoned 0xAA by harness; must re-init every call) ----------
__global__ void k_init(unsigned* m1, int n1, unsigned* m2, int n2,
                       float* s1, float* wsum1, int ns1,
                       float* s2, int ns2, float* acc2, int nacc) {
    int i = blockIdx.x * blockDim.x + threadIdx.x;
    if (i < n1)  m1[i] = FORD_NEG_INF;
    if (i < n2)  m2[i] = FORD_NEG_INF;
    if (i < ns1) { s1[i] = 0.f; wsum1[i] = 0.f; }
    if (i < ns2) s2[i] = 0.f;
    if (i < nacc) acc2[i] = 0.f;
}

// ---------- layer 1: per-node attention logits (IN=1 => logits = x[n] * (W1 . a)) ----------
__global__ void k_node1(const float* __restrict__ x, const float* __restrict__ W1,
                        const float* __restrict__ as1, const float* __restrict__ ad1,
                        float* __restrict__ alS1, float* __restrict__ alD1, int N) {
    int n = blockIdx.x * blockDim.x + threadIdx.x;
    if (n >= N) return;
    float xv = x[n];
#pragma unroll
    for (int h = 0; h < 4; ++h) {
        float ws = 0.f, wd = 0.f;
#pragma unroll
        for (int c = 0; c < 8; ++c) {
            float w = W1[h * 8 + c];
            ws += w * as1[h * 8 + c];
            wd += w * ad1[h * 8 + c];
        }
        alS1[n * 4 + h] = xv * ws;
        alD1[n * 4 + h] = xv * wd;
    }
}

// ---------- layer 1: edge pass A — segment max ----------
__global__ void k_edge1_max(const int* __restrict__ ei, int E, int N,
                            const float* __restrict__ alS1, const float* __restrict__ alD1,
                            unsigned* __restrict__ m1) {
    int e = blockIdx.x * blockDim.x + threadIdx.x;
    if (e >= E + N) return;
    int s, d;
    if (e < E) { s = ei[e]; d = ei[E + e]; } else { s = d = e - E; }
    float4 as = *reinterpret_cast<const float4*>(alS1 + (size_t)s * 4);
    float4 ad = *reinterpret_cast<const float4*>(alD1 + (size_t)d * 4);
    atomicMax(m1 + (size_t)d * 4 + 0, ford(lrelu(as.x + ad.x)));
    atomicMax(m1 + (size_t)d * 4 + 1, ford(lrelu(as.y + ad.y)));
    atomicMax(m1 + (size_t)d * 4 + 2, ford(lrelu(as.z + ad.z)));
    atomicMax(m1 + (size_t)d * 4 + 3, ford(lrelu(as.w + ad.w)));
}

// ---------- layer 1: edge pass B — exp + factorized accumulation (4 floats/edge, not 32) ----------
__global__ void k_edge1_sum(const int* __restrict__ ei, int E, int N,
                            const float* __restrict__ x,
                            const float* __restrict__ alS1, const float* __restrict__ alD1,
                            const unsigned* __restrict__ m1,
                            float* __restrict__ s1, float* __restrict__ wsum1) {
    int e = blockIdx.x * blockDim.x + threadIdx.x;
    if (e >= E + N) return;
    int s, d;
    if (e < E) { s = ei[e]; d = ei[E + e]; } else { s = d = e - E; }
    float4 as = *reinterpret_cast<const float4*>(alS1 + (size_t)s * 4);
    float4 ad = *reinterpret_cast<const float4*>(alD1 + (size_t)d * 4);
    uint4  mm = *reinterpret_cast<const uint4*>(m1 + (size_t)d * 4);
    float xs = x[s];
    float t0 = __expf(lrelu(as.x + ad.x) - forddec(mm.x));
    float t1 = __expf(lrelu(as.y + ad.y) - forddec(mm.y));
    float t2 = __expf(lrelu(as.z + ad.z) - forddec(mm.z));
    float t3 = __expf(lrelu(as.w + ad.w) - forddec(mm.w));
    atomicAdd(s1 + (size_t)d * 4 + 0, t0);
    atomicAdd(s1 + (size_t)d * 4 + 1, t1);
    atomicAdd(s1 + (size_t)d * 4 + 2, t2);
    atomicAdd(s1 + (size_t)d * 4 + 3, t3);
    atomicAdd(wsum1 + (size_t)d * 4 + 0, t0 * xs);
    atomicAdd(wsum1 + (size_t)d * 4 + 1, t1 * xs);
    atomicAdd(wsum1 + (size_t)d * 4 + 2, t2 * xs);
    atomicAdd(wsum1 + (size_t)d * 4 + 3, t3 * xs);
}

// ---------- layer 1 finalize: out1[n,h*8+c] = relu(W1[h,c]*wsum/s + b1) ----------
__global__ void k_fin1(const float* __restrict__ W1, const float* __restrict__ b1,
                       const float* __restrict__ s1, const float* __restrict__ wsum1,
                       float* __restrict__ out1, int N) {
    int i = blockIdx.x * blockDim.x + threadIdx.x;
    if (i >= N * 32) return;
    int n = i >> 5, j = i & 31, h = j >> 3;
    float v = W1[j] * (wsum1[n * 4 + h] / s1[n * 4 + h]) + b1[j];
    out1[i] = v > 0.f ? v : 0.f;
}

// ---------- layer 2 transform via WMMA f32 16x16x4 chain (K=32) ----------
// B is W2 zero-padded to [32,16]; col 8 = W2·a_src2, col 9 = W2·a_dst2 so the
// attention logits come out of the same WMMA as h2. One wave = one 16-node tile.
__global__ void k_gemm2(const float* __restrict__ out1, const float* __restrict__ W2,
                        const float* __restrict__ as2, const float* __restrict__ ad2,
                        float* __restrict__ h2, float* __restrict__ alS2,
                        float* __restrict__ alD2, int N) {
    __shared__ float Bp[32 * 16];
    int t = threadIdx.x;
    for (int i = t; i < 512; i += blockDim.x) {
        int k = i >> 4, n = i & 15;
        float v = 0.f;
        if (n < 8) {
            v = W2[k * 8 + n];
        } else if (n == 8) {
            for (int c = 0; c < 8; ++c) v += W2[k * 8 + c] * as2[c];
        } else if (n == 9) {
            for (int c = 0; c < 8; ++c) v += W2[k * 8 + c] * ad2[c];
        }
        Bp[i] = v;
    }
    __syncthreads();

    int wave = t >> 5, lane = t & 31;
    long tile = (long)blockIdx.x * 8 + wave;   // 8 waves per block
    long row0 = tile * 16;
    if (row0 >= N) return;                      // wave-uniform exit

    int m = lane & 15, khi = lane >> 4;
    long rowm = row0 + m;
    if (rowm >= N) rowm = N - 1;                // clamp loads: keep EXEC all-1s for WMMA
    const float* Arow = out1 + rowm * 32;

    v8f c = {};
#pragma unroll
    for (int step = 0; step < 8; ++step) {
        int k0 = step * 4 + khi * 2;
        v2f a; a.x = Arow[k0];          a.y = Arow[k0 + 1];
        v2f b; b.x = Bp[k0 * 16 + m];   b.y = Bp[(k0 + 1) * 16 + m];
        // (neg_a, A, neg_b, B, c_mod, C, reuse_a, reuse_b)
        c = __builtin_amdgcn_wmma_f32_16x16x4_f32(false, a, false, b, (short)0, c,
                                                  false, false);
    }

    // D layout: lane col = lane&15; VGPR r holds row M = 8*khi + r
#pragma unroll
    for (int r = 0; r < 8; ++r) {
        long node = row0 + khi * 8 + r;
        if (node >= N) continue;
        float v = c[r];
        if (m < 8)        h2[node * 8 + m] = v;
        else if (m == 8)  alS2[node] = v;
        else if (m == 9)  alD2[node] = v;
    }
}

// ---------- layer 2: edge pass A — segment max ----------
__global__ void k_edge2_max(const int* __restrict__ ei, int E, int N,
                            const float* __restrict__ alS2, const float* __restrict__ alD2,
                            unsigned* __restrict__ m2) {
    int e = blockIdx.x * blockDim.x + threadIdx.x;
    if (e >= E + N) return;
    int s, d;
    if (e < E) { s = ei[e]; d = ei[E + e]; } else { s = d = e - E; }
    atomicMax(m2 + d, ford(lrelu(alS2[s] + alD2[d])));
}

// ---------- layer 2: edge pass B — exp + weighted aggregation ----------
__global__ void k_edge2_sum(const int* __restrict__ ei, int E, int N,
                            const float* __restrict__ h2,
                            const float* __restrict__ alS2, const float* __restrict__ alD2,
                            const unsigned* __restrict__ m2,
                            float* __restrict__ s2, float* __restrict__ acc2) {
    int e = blockIdx.x * blockDim.x + threadIdx.x;
    if (e >= E + N) return;
    int s, d;
    if (e < E) { s = ei[e]; d = ei[E + e]; } else { s = d = e - E; }
    float t = __expf(lrelu(alS2[s] + alD2[d]) - forddec(m2[d]));
    atomicAdd(s2 + d, t);
    float4 h0 = *reinterpret_cast<const float4*>(h2 + (size_t)s * 8);
    float4 h1 = *reinterpret_cast<const float4*>(h2 + (size_t)s * 8 + 4);
    float* ad = acc2 + (size_t)d * 8;
    atomicAdd(ad + 0, t * h0.x); atomicAdd(ad + 1, t * h0.y);
    atomicAdd(ad + 2, t * h0.z); atomicAdd(ad + 3, t * h0.w);
    atomicAdd(ad + 4, t * h1.x); atomicAdd(ad + 5, t * h1.y);
    atomicAdd(ad + 6, t * h1.z); atomicAdd(ad + 7, t * h1.w);
}

// ---------- layer 2 finalize ----------
__global__ void k_fin2(const float* __restrict__ acc2, const float* __restrict__ s2,
                       const float* __restrict__ b2, float* __restrict__ out, int N) {
    int i = blockIdx.x * blockDim.x + threadIdx.x;
    if (i >= N * 8) return;
    int n = i >> 3, c = i & 7;
    out[i] = acc2[i] / s2[n] + b2[c];
}

extern "C" void kernel_launch(void* const* d_in, const int* in_sizes, int n_in,
                              void* d_out, int out_size, void* d_ws, size_t ws_size,
                              hipStream_t stream) {
    const float* x   = (const float*)d_in[0];
    const int*   ei  = (const int*)d_in[1];
    const float* W1  = (const float*)d_in[2];
    const float* as1 = (const float*)d_in[3];
    const float* ad1 = (const float*)d_in[4];
    const float* b1  = (const float*)d_in[5];
    const float* W2  = (const float*)d_in[6];
    const float* as2 = (const float*)d_in[7];
    const float* ad2 = (const float*)d_in[8];
    const float* b2  = (const float*)d_in[9];
    (void)n_in; (void)out_size; (void)ws_size;

    int N = in_sizes[0];       // IN == 1, so |x| == N
    int E = in_sizes[1] / 2;   // edge_index is [2, E]
    int EN = E + N;            // edges + implicit self-loops

    // workspace carve (256B-aligned partitions; ~29 MB total)
    char* ws = (char*)d_ws;
    size_t cur = 0;
    auto carve = [&](size_t bytes) -> void* {
        void* p = ws + cur;
        cur += (bytes + 255) & ~(size_t)255;
        return p;
    };
    unsigned* m1    = (unsigned*)carve(sizeof(unsigned) * (size_t)N * 4);
    unsigned* m2    = (unsigned*)carve(sizeof(unsigned) * (size_t)N);
    float*    s1    = (float*)carve(sizeof(float) * (size_t)N * 4);
    float*    wsum1 = (float*)carve(sizeof(float) * (size_t)N * 4);
    float*    s2    = (float*)carve(sizeof(float) * (size_t)N);
    float*    acc2  = (float*)carve(sizeof(float) * (size_t)N * 8);
    float*    alS1  = (float*)carve(sizeof(float) * (size_t)N * 4);
    float*    alD1  = (float*)carve(sizeof(float) * (size_t)N * 4);
    float*    out1  = (float*)carve(sizeof(float) * (size_t)N * 32);
    float*    h2    = (float*)carve(sizeof(float) * (size_t)N * 8);
    float*    alS2  = (float*)carve(sizeof(float) * (size_t)N);
    float*    alD2  = (float*)carve(sizeof(float) * (size_t)N);

    const int B = 256;
    k_init<<<(8 * N + B - 1) / B, B, 0, stream>>>(m1, 4 * N, m2, N,
                                                  s1, wsum1, 4 * N, s2, N, acc2, 8 * N);
    k_node1<<<(N + B - 1) / B, B, 0, stream>>>(x, W1, as1, ad1, alS1, alD1, N);
    k_edge1_max<<<(EN + B - 1) / B, B, 0, stream>>>(ei, E, N, alS1, alD1, m1);
    k_edge1_sum<<<(EN + B - 1) / B, B, 0, stream>>>(ei, E, N, x, alS1, alD1, m1, s1, wsum1);
    k_fin1<<<(32 * N + B - 1) / B, B, 0, stream>>>(W1, b1, s1, wsum1, out1, N);

    int tiles = (N + 15) / 16;
    k_gemm2<<<(tiles + 7) / 8, B, 0, stream>>>(out1, W2, as2, ad2, h2, alS2, alD2, N);

    k_edge2_max<<<(EN + B - 1) / B, B, 0, stream>>>(ei, E, N, alS2, alD2, m2);
    k_edge2_sum<<<(EN + B - 1) / B, B, 0, stream>>>(ei, E, N, h2, alS2, alD2, m2, s2, acc2);
    k_fin2<<<(8 * N + B - 1) / B, B, 0, stream>>>(acc2, s2, b2, (float*)d_out, N);
}